// Metalayer_20272245637606
// MI455X (gfx1250) — compile-verified
//
#include <hip/hip_runtime.h>

// ---------------------------------------------------------------------------
// Metalayer (coupled-mode photonic layer) for gfx1250 / MI455X.
// All dense matmuls (MLP hidden layers, Pade-13 expm polynomial chain,
// complex squarings) run on V_WMMA_F32_16X16X4_F32 (wave32, 16x16 f32 tile).
// The near-singular 768x768 coupling solve and the Pade denominator solve use
// exact Gauss-Jordan (graph-replayed pivot loop). The complex Pade solve is
// reduced to a real solve via commutativity of V and W (both polynomials in G):
//   P = (V - iW)^-1 (V + iW) = (V^2 + W^2)^-1 [(V^2 - W^2) + 2iVW].
// ---------------------------------------------------------------------------

#if defined(__HIP_DEVICE_COMPILE__)
#if !__has_builtin(__builtin_amdgcn_wmma_f32_16x16x4_f32)
#error "gfx1250 f32 WMMA builtin __builtin_amdgcn_wmma_f32_16x16x4_f32 not available"
#endif
#endif

typedef float v2f __attribute__((ext_vector_type(2)));
typedef float v8f __attribute__((ext_vector_type(8)));

#define NWG   384
#define MODES 2
#define RES   128
#define NBR   6
#define EY    768           // NBR*RES
#define MDIM  768           // NWG*MODES
#define NEDGE 2304          // NWG*NBR
#define E0LEN 49920         // (NWG+NBR)*RES
#define PADL  320           // ((2*KNN+1)*RES)//2
#define OUTN  49152         // NWG*RES
#define KVEC  7.853981633974483f   // 2*pi/0.8
#define HMIN  0.075f
#define HMAX  0.225f

// ------------------------------ WMMA GEMM ----------------------------------
// Row-major tight GEMM: D = alpha*A@B + beta*D [+ bias[col]] [+ diagAdd on i==j],
// optional relu. One wave computes one 16x16 tile; K stepped by 4.
// A-frag layout (ISA 7.12.2): lane m=lane&15, k = k0 + (lane>>4)*2 + {0,1}.
// B-frag: lane n=lane&15, same K split. C/D: vgpr v -> (M=(lane>>4)*8+v, N=lane&15).
__global__ __launch_bounds__(256)
void k_gemm_f32_wmma(const float* __restrict__ A, const float* __restrict__ B,
                     float* __restrict__ D, const float* __restrict__ bias,
                     int M, int N, int K, float alpha, float beta,
                     float diagAdd, int relu)
{
    const int lane  = threadIdx.x & 31;
    const int wave  = threadIdx.x >> 5;
    const int tilesN = N >> 4;
    const int tiles  = (M >> 4) * tilesN;
    const int tile   = blockIdx.x * (blockDim.x >> 5) + wave;
    if (tile >= tiles) return;                       // whole-wave uniform exit
    const int row0 = (tile / tilesN) << 4;
    const int col0 = (tile % tilesN) << 4;

    const int ma   = lane & 15;                      // A row / B,C col in tile
    const int koff = (lane >> 4) << 1;               // 0 or 2

    v8f acc = {0.f, 0.f, 0.f, 0.f, 0.f, 0.f, 0.f, 0.f};
    const float* Arow = A + (size_t)(row0 + ma) * K;
    const float* Bcol = B + col0 + ma;
    for (int k = 0; k < K; k += 4) {
        v2f a, b;
        a.x = Arow[k + koff];
        a.y = Arow[k + koff + 1];
        b.x = Bcol[(size_t)(k + koff) * N];
        b.y = Bcol[(size_t)(k + koff + 1) * N];
#if defined(__HIP_DEVICE_COMPILE__)
        acc = __builtin_amdgcn_wmma_f32_16x16x4_f32(
            /*neg_a=*/false, a, /*neg_b=*/false, b,
            /*c_mod=*/(short)0, acc, /*reuse_a=*/false, /*reuse_b=*/false);
#endif
    }

    const int mbase = (lane >> 4) << 3;              // 0 or 8
    #pragma unroll
    for (int v = 0; v < 8; ++v) {
        const int r = row0 + mbase + v;
        const int c = col0 + ma;
        float val = alpha * acc[v];
        if (beta != 0.f) val += beta * D[(size_t)r * N + c];
        if (bias)        val += bias[c];
        if (r == c)      val += diagAdd;
        if (relu)        val = fmaxf(val, 0.f);
        D[(size_t)r * N + c] = val;
    }
}

// ------------------------------ small kernels ------------------------------
__global__ void k_hs(const float* __restrict__ hp, float* __restrict__ hs, int n)
{
    int i = blockIdx.x * blockDim.x + threadIdx.x;
    if (i >= n) return;
    float s = 1.f / (1.f + expf(-hp[i]));
    hs[i] = s * (HMAX - HMIN) + HMIN;
}

// hidden layer 1 from a scalar feature: out[n,j] = relu(x[n]*W[j] + b[j])
__global__ void k_layer1(const float* __restrict__ x, const float* __restrict__ W,
                         const float* __restrict__ b, float* __restrict__ out,
                         int rows, int H)
{
    int t = blockIdx.x * blockDim.x + threadIdx.x;
    if (t >= rows * H) return;
    int n = t / H, j = t - n * H;
    out[t] = fmaxf(x[n] * W[j] + b[j], 0.f);
}

// generic small dense: out[r,o] = (relu)(X[r,:K] @ W[:,o] + b[o])
__global__ void k_dense_small(const float* __restrict__ X, const float* __restrict__ W,
                              const float* __restrict__ b, float* __restrict__ out,
                              int rows, int K, int O, int relu)
{
    int t = blockIdx.x * blockDim.x + threadIdx.x;
    if (t >= rows * O) return;
    int r = t / O, o = t - r * O;
    float s = b[o];
    const float* xr = X + (size_t)r * K;
    for (int k = 0; k < K; ++k) s += xr[k] * W[(size_t)k * O + o];
    out[t] = relu ? fmaxf(s, 0.f) : s;
}

// gen_U0: U0[n*2+m] = 2*eta0*DX * sum_j Ey[n,m,j]*E0p[n*128+j]
__global__ void k_u0(const float* __restrict__ E0, const float* __restrict__ Ey,
                     const float* __restrict__ neffs, float* __restrict__ U0)
{
    int i = blockIdx.x * blockDim.x + threadIdx.x;
    if (i >= MDIM) return;
    int n = i >> 1;
    const float* ey = Ey + (size_t)i * EY;
    float s = 0.f;
    for (int j = 0; j < EY; ++j) {
        int t = n * RES + j;                         // index into padded E0
        float e0 = (t >= PADL && t < PADL + E0LEN) ? E0[t - PADL] : 0.f;
        s += ey[j] * e0;
    }
    float ne = neffs[i];
    float eta0 = ne * 1.46f / (ne + 1.46f);
    U0[i] = 2.f * eta0 * s * 0.01f;                  // 2*C_EPS*eta0*E_sum*DX
}

// edge features: [hs[src], hs[dst], offset]
__global__ void k_ck_inputs(const float* __restrict__ hs, float* __restrict__ CK)
{
    int e = blockIdx.x * blockDim.x + threadIdx.x;
    if (e >= NEDGE) return;
    int n = e / NBR, o = e - n * NBR;
    int off = o - 3;                                 // offsets -3..2
    int dst = (n + off + NWG) % NWG;
    CK[e * 3 + 0] = hs[n];
    CK[e * 3 + 1] = hs[dst];
    CK[e * 3 + 2] = (float)off;
}

// scatter edge 2x2 mode blocks into dense C,K (indices unique; plain stores)
__global__ void k_scatter(const float* __restrict__ Cv, const float* __restrict__ Kv,
                          float* __restrict__ C, float* __restrict__ K)
{
    int t = blockIdx.x * blockDim.x + threadIdx.x;
    if (t >= NEDGE * 4) return;
    int e = t >> 2, q = t & 3;
    int mi = q >> 1, mj = q & 1;
    int n = e / NBR, off = (e - n * NBR) - 3;
    int dst = (n + off + NWG) % NWG;
    int row = n * MODES + mi, col = dst * MODES + mj;
    C[(size_t)row * MDIM + col] = Cv[e * 4 + q];
    K[(size_t)row * MDIM + col] = Kv[e * 4 + q];
}

// M1 = diag(neffs*k) @ C + K   (row scale)
__global__ void k_eigprep(const float* __restrict__ C, const float* __restrict__ K,
                          const float* __restrict__ neffs, float* __restrict__ M1)
{
    int t = blockIdx.x * blockDim.x + threadIdx.x;
    if (t >= MDIM * MDIM) return;
    int i = t / MDIM;
    M1[t] = neffs[i] * KVEC * C[t] + K[t];
}

// out = (accum?out:0) + ca*X + cb*Y + cc*Z (+dadd on diagonal of n x n)
__global__ void k_axpy3(float* __restrict__ out, const float* __restrict__ X, float ca,
                        const float* __restrict__ Y, float cb,
                        const float* __restrict__ Z, float cc,
                        int n, float dadd, int accum)
{
    int t = blockIdx.x * blockDim.x + threadIdx.x;
    if (t >= n * n) return;
    float v = accum ? out[t] : 0.f;
    if (X) v += ca * X[t];
    if (Y) v += cb * Y[t];
    if (Z) v += cc * Z[t];
    if (dadd != 0.f) { int i = t / n; if (t - i * n == i) v += dadd; }
    out[t] = v;
}

// S = VV+WW (into VV slot); Nr = VV-WW (into WW slot)
__global__ void k_sumdiff(float* __restrict__ VV, float* __restrict__ WW, int total)
{
    int t = blockIdx.x * blockDim.x + threadIdx.x;
    if (t >= total) return;
    float a = VV[t], b = WW[t];
    VV[t] = a + b;
    WW[t] = a - b;
}

// -------- Gauss-Jordan (no pivot search; diagonal pivoting) ----------------
// prep: single workgroup -> no cross-block races. Saves multiplier column,
// then normalizes pivot row of A and of the RHS blocks.
__global__ __launch_bounds__(1024)
void k_gj_prep(float* __restrict__ A, float* __restrict__ R0, float* __restrict__ R1,
               float* __restrict__ mcol, int p, int n)
{
    __shared__ float rinv;
    int tid = threadIdx.x;
    if (tid == 0) rinv = 1.0f / A[(size_t)p * n + p];
    __syncthreads();
    for (int i = tid; i < n; i += blockDim.x) mcol[i] = A[(size_t)i * n + p];
    __syncthreads();
    for (int j = tid; j < n; j += blockDim.x) A[(size_t)p * n + j] *= rinv;
    for (int j = tid; j < n; j += blockDim.x) R0[(size_t)p * n + j] *= rinv;
    if (R1)
        for (int j = tid; j < n; j += blockDim.x) R1[(size_t)p * n + j] *= rinv;
}

// eliminate: row i (!=p), width W = n + n*(R1?2:1). Reads only pivot row + mcol.
__global__ void k_gj_elim(float* __restrict__ A, float* __restrict__ R0,
                          float* __restrict__ R1, const float* __restrict__ mcol,
                          int p, int n, int W)
{
    int t = blockIdx.x * blockDim.x + threadIdx.x;
    if (t >= n * W) return;
    int i = t / W;
    if (i == p) return;
    int j = t - i * W;
    float f = mcol[i];
    if (j < n) {
        A[(size_t)i * n + j] -= f * A[(size_t)p * n + j];
    } else {
        j -= n;
        if (j < n) R0[(size_t)i * n + j] -= f * R0[(size_t)p * n + j];
        else { j -= n; R1[(size_t)i * n + j] -= f * R1[(size_t)p * n + j]; }
    }
}

// Uz = (Xr + iXi) @ U0 (U0 real)
__global__ void k_matvec2(const float* __restrict__ Xr, const float* __restrict__ Xi,
                          const float* __restrict__ U0,
                          float* __restrict__ Uzr, float* __restrict__ Uzi)
{
    int i = blockIdx.x * blockDim.x + threadIdx.x;
    if (i >= MDIM) return;
    float sr = 0.f, si = 0.f;
    const float* xr = Xr + (size_t)i * MDIM;
    const float* xi = Xi + (size_t)i * MDIM;
    for (int j = 0; j < MDIM; ++j) { float u = U0[j]; sr += xr[j] * u; si += xi[j] * u; }
    Uzr[i] = sr;
    Uzi[i] = si;
}

// gen_En gather: each output sample sums the <=6 overlapping windows x 2 modes.
__global__ void k_gen_en(const float* __restrict__ Ey, const float* __restrict__ neffs,
                         const float* __restrict__ Uzr, const float* __restrict__ Uzi,
                         float* __restrict__ out)
{
    int g = blockIdx.x * blockDim.x + threadIdx.x;
    if (g >= OUTN) return;
    int p = g + PADL;
    int nhi = p >> 7;               if (nhi > NWG - 1) nhi = NWG - 1;
    int nlo = (p >= EY) ? ((p - (EY - 1) + (RES - 1)) >> 7) : 0;
    float re = 0.f, im = 0.f;
    for (int n = nlo; n <= nhi; ++n) {
        int j = p - n * RES;                          // 0..EY-1
        #pragma unroll
        for (int m = 0; m < MODES; ++m) {
            int idx = n * MODES + m;
            float ne  = neffs[idx];
            float eta = ne / (ne + 1.f);              // n0 = 1
            float ey  = Ey[(size_t)idx * EY + j];
            re += eta * Uzr[idx] * ey;
            im += eta * Uzi[idx] * ey;
        }
    }
    out[2 * g + 0] = re;                              // interleaved complex64
    out[2 * g + 1] = im;
}

// ------------------------------- host side ---------------------------------
extern "C" void kernel_launch(void* const* d_in, const int* in_sizes, int n_in,
                              void* d_out, int out_size, void* d_ws, size_t ws_size,
                              hipStream_t stream)
{
    (void)in_sizes; (void)n_in; (void)out_size; (void)ws_size;

    // pytree-flattened inputs: E0, h_paras, then per MLP (W1,W2,W3,b1,b2,b3)
    const float* E0   = (const float*)d_in[0];
    const float* hpar = (const float*)d_in[1];
    const float* nW1 = (const float*)d_in[2],  *nW2 = (const float*)d_in[3],
               * nW3 = (const float*)d_in[4],  *nb1 = (const float*)d_in[5],
               * nb2 = (const float*)d_in[6],  *nb3 = (const float*)d_in[7];
    const float* eW1 = (const float*)d_in[8],  *eW2 = (const float*)d_in[9],
               * eW3 = (const float*)d_in[10], *eb1 = (const float*)d_in[11],
               * eb2 = (const float*)d_in[12], *eb3 = (const float*)d_in[13];
    const float* cW1 = (const float*)d_in[14], *cW2 = (const float*)d_in[15],
               * cW3 = (const float*)d_in[16], *cb1 = (const float*)d_in[17],
               * cb2 = (const float*)d_in[18], *cb3 = (const float*)d_in[19];
    const float* kW1 = (const float*)d_in[20], *kW2 = (const float*)d_in[21],
               * kW3 = (const float*)d_in[22], *kb1 = (const float*)d_in[23],
               * kb2 = (const float*)d_in[24], *kb3 = (const float*)d_in[25];

    // -------- workspace carve-up (~21.5 MB of f32) --------
    float* w = (float*)d_ws;
    size_t off = 0;
    auto alloc = [&](size_t nfl) { float* p = w + off; off += nfl; return p; };
    float* hs   = alloc(NWG);
    float* neff = alloc(MDIM);
    float* h1   = alloc((size_t)NWG * 256);
    float* h2   = alloc((size_t)NWG * 256);
    float* U0   = alloc(MDIM);
    float* CKf  = alloc((size_t)NEDGE * 3);
    float* EH1  = alloc((size_t)NEDGE * 64);
    float* EH2  = alloc((size_t)NEDGE * 64);
    float* Cv   = alloc((size_t)NEDGE * 4);
    float* Kv   = alloc((size_t)NEDGE * 4);
    float* Uzr  = alloc(MDIM);
    float* Uzi  = alloc(MDIM);
    float* mcol = alloc(MDIM);
    float* Ey   = alloc((size_t)NWG * MODES * EY);     // persistent
    const size_t MM = (size_t)MDIM * MDIM;
    float* P[7];
    for (int i = 0; i < 7; ++i) P[i] = alloc(MM);

    auto nblk = [](int n, int b) { return (n + b - 1) / b; };
    auto gemm = [&](const float* A, const float* B, float* D, int M, int N, int K,
                    float alpha, float beta, const float* bias, float diagAdd, int relu) {
        int tiles = (M / 16) * (N / 16);
        k_gemm_f32_wmma<<<nblk(tiles, 8), 256, 0, stream>>>(
            A, B, D, bias, M, N, K, alpha, beta, diagAdd, relu);
    };
    auto axpy3 = [&](float* out, const float* X, float ca, const float* Y, float cb,
                     const float* Z, float cc, float dadd, int accum) {
        k_axpy3<<<nblk((int)MM, 256), 256, 0, stream>>>(out, X, ca, Y, cb, Z, cc,
                                                        MDIM, dadd, accum);
    };
    auto gj_solve = [&](float* A, float* R0, float* R1) {
        int W = MDIM + MDIM * (R1 ? 2 : 1);
        int grid = nblk(MDIM * W, 256);
        for (int p = 0; p < MDIM; ++p) {
            k_gj_prep<<<1, 1024, 0, stream>>>(A, R0, R1, mcol, p, MDIM);
            k_gj_elim<<<grid, 256, 0, stream>>>(A, R0, R1, mcol, p, MDIM, W);
        }
    };

    // Pade-13 coefficients normalized by b0 (scale-invariant in the solve)
    static const double bp[14] = {
        64764752532480000., 32382376266240000., 7771770303897600.,
        1187353796428800., 129060195264000., 10559470521600.,
        670442572800., 33522128640., 1323241920., 40840800.,
        960960., 16380., 182., 1. };
    float c[14];
    for (int i = 0; i < 14; ++i) c[i] = (float)(bp[i] / bp[0]);
    const float gscale = 0.6f / 16.f;                 // wh * 2^-SQUARINGS

    // -------- 1) hs, neff MLP --------
    k_hs<<<nblk(NWG, 256), 256, 0, stream>>>(hpar, hs, NWG);
    k_layer1<<<nblk(NWG * 64, 256), 256, 0, stream>>>(hs, nW1, nb1, h1, NWG, 64);
    gemm(h1, nW2, h2, NWG, 64, 64, 1.f, 0.f, nb2, 0.f, 1);
    k_dense_small<<<nblk(NWG * MODES, 256), 256, 0, stream>>>(h2, nW3, nb3, neff,
                                                              NWG, 64, MODES, 0);
    // -------- 2) enn MLP -> Ey, then U0 --------
    k_layer1<<<nblk(NWG * 256, 256), 256, 0, stream>>>(hs, eW1, eb1, h1, NWG, 256);
    gemm(h1, eW2, h2, NWG, 256, 256, 1.f, 0.f, eb2, 0.f, 1);
    gemm(h2, eW3, Ey, NWG, MODES * EY, 256, 1.f, 0.f, eb3, 0.f, 0);
    k_u0<<<nblk(MDIM, 256), 256, 0, stream>>>(E0, Ey, neff, U0);

    // -------- 3) edge MLPs -> C, K --------
    k_ck_inputs<<<nblk(NEDGE, 256), 256, 0, stream>>>(hs, CKf);
    k_dense_small<<<nblk(NEDGE * 64, 256), 256, 0, stream>>>(CKf, cW1, cb1, EH1,
                                                             NEDGE, 3, 64, 1);
    gemm(EH1, cW2, EH2, NEDGE, 64, 64, 1.f, 0.f, cb2, 0.f, 1);
    k_dense_small<<<nblk(NEDGE * 4, 256), 256, 0, stream>>>(EH2, cW3, cb3, Cv,
                                                            NEDGE, 64, 4, 0);
    k_dense_small<<<nblk(NEDGE * 64, 256), 256, 0, stream>>>(CKf, kW1, kb1, EH1,
                                                             NEDGE, 3, 64, 1);
    gemm(EH1, kW2, EH2, NEDGE, 64, 64, 1.f, 0.f, kb2, 0.f, 1);
    k_dense_small<<<nblk(NEDGE * 4, 256), 256, 0, stream>>>(EH2, kW3, kb3, Kv,
                                                            NEDGE, 64, 4, 0);
    (void)hipMemsetAsync(P[0], 0, MM * sizeof(float), stream);   // C
    (void)hipMemsetAsync(P[1], 0, MM * sizeof(float), stream);   // K
    k_scatter<<<nblk(NEDGE * 4, 256), 256, 0, stream>>>(Cv, Kv, P[0], P[1]);

    // -------- 4) EigM = C^-1 (B C + K), G = (wh/16) EigM --------
    k_eigprep<<<nblk((int)MM, 256), 256, 0, stream>>>(P[0], P[1], neff, P[2]);
    gj_solve(P[0], P[2], nullptr);                    // P2 <- EigM (P0 destroyed)
    axpy3(P[1], P[2], gscale, nullptr, 0.f, nullptr, 0.f, 0.f, 0);  // G = P1

    // -------- 5) Pade-13 polynomial chain (A = iG) --------
    float* G = P[1];
    gemm(G, G, P[0], MDIM, MDIM, MDIM, -1.f, 0.f, nullptr, 0.f, 0);   // A2 = -G^2
    gemm(P[0], P[0], P[2], MDIM, MDIM, MDIM, 1.f, 0.f, nullptr, 0.f, 0); // A4
    gemm(P[0], P[2], P[3], MDIM, MDIM, MDIM, 1.f, 0.f, nullptr, 0.f, 0); // A6
    float *A2 = P[0], *A4 = P[2], *A6 = P[3], *T = P[4];
    axpy3(T, A6, c[13], A4, c[11], A2, c[9], 0.f, 0);
    gemm(A6, T, P[5], MDIM, MDIM, MDIM, 1.f, 0.f, nullptr, 0.f, 0);   // Uin
    axpy3(P[5], A6, c[7], A4, c[5], A2, c[3], c[1], 1);
    gemm(G, P[5], P[6], MDIM, MDIM, MDIM, 1.f, 0.f, nullptr, 0.f, 0); // W (U = iW)
    axpy3(T, A6, c[12], A4, c[10], A2, c[8], 0.f, 0);
    gemm(A6, T, P[5], MDIM, MDIM, MDIM, 1.f, 0.f, nullptr, 0.f, 0);   // V
    axpy3(P[5], A6, c[6], A4, c[4], A2, c[2], c[0], 1);
    float *V = P[5], *Wm = P[6];

    // -------- 6) (V-iW)^-1(V+iW) = (V^2+W^2)^-1[(V^2-W^2) + 2iVW] --------
    gemm(V,  V,  P[0], MDIM, MDIM, MDIM, 1.f, 0.f, nullptr, 0.f, 0);  // VV
    gemm(Wm, Wm, P[1], MDIM, MDIM, MDIM, 1.f, 0.f, nullptr, 0.f, 0);  // WW
    gemm(V,  Wm, P[2], MDIM, MDIM, MDIM, 2.f, 0.f, nullptr, 0.f, 0);  // Ni = 2VW
    k_sumdiff<<<nblk((int)MM, 256), 256, 0, stream>>>(P[0], P[1], (int)MM);
    gj_solve(P[0], P[1], P[2]);                       // P1 <- Xr, P2 <- Xi

    // -------- 7) 4 squarings of X = Xr + iXi --------
    float *sr = P[1], *si = P[2], *tr = P[3], *ti = P[4];
    for (int s = 0; s < 4; ++s) {
        gemm(sr, sr, tr, MDIM, MDIM, MDIM, 1.f, 0.f, nullptr, 0.f, 0);
        gemm(si, si, tr, MDIM, MDIM, MDIM, -1.f, 1.f, nullptr, 0.f, 0);
        gemm(sr, si, ti, MDIM, MDIM, MDIM, 1.f, 0.f, nullptr, 0.f, 0);
        gemm(si, sr, ti, MDIM, MDIM, MDIM, 1.f, 1.f, nullptr, 0.f, 0);
        float* t0 = sr; sr = tr; tr = t0;
        float* t1 = si; si = ti; ti = t1;
    }

    // -------- 8) Uz = P @ U0 ; gen_En overlap-add (gather form) --------
    k_matvec2<<<nblk(MDIM, 256), 256, 0, stream>>>(sr, si, U0, Uzr, Uzi);
    k_gen_en<<<nblk(OUTN, 256), 256, 0, stream>>>(Ey, neff, Uzr, Uzi, (float*)d_out);
}